// HyperGraph_Class_70403103916536
// MI455X (gfx1250) — compile-verified
//
#include <hip/hip_runtime.h>
#include <hip/hip_bf16.h>
#include <math.h>

// Problem constants (match reference)
#define NN 100000
#define NF 128
#define NHE 50000
#define NE 1000000
#define H 8
#define C 8
#define HC 64       // H*C
#define C2 8

typedef __attribute__((ext_vector_type(2))) float v2f;
typedef __attribute__((ext_vector_type(8))) float v8f;

// ---------- order-preserving float <-> uint encoding for atomic segment-max ----------
__device__ __forceinline__ unsigned encf(float x) {
    unsigned u = __float_as_uint(x);
    return (u >> 31) ? ~u : (u | 0x80000000u);
}
__device__ __forceinline__ float decf(unsigned e) {
    unsigned u = (e >> 31) ? (e & 0x7FFFFFFFu) : ~e;
    return __uint_as_float(u);
}
#define ENC_NEGINF 0x007FFFFFu   // encf(-inf)

// ---------- fills ----------
__global__ void k_fill_f(float* p, float v, int n) {
    int i = blockIdx.x * 256 + threadIdx.x;
    if (i < n) p[i] = v;
}
__global__ void k_fill_u(unsigned* p, unsigned v, int n) {
    int i = blockIdx.x * 256 + threadIdx.x;
    if (i < n) p[i] = v;
}

// ---------- degrees ----------
__global__ void k_degree(const int* __restrict__ ni, const int* __restrict__ hi,
                         float* degN, float* degHE) {
    int e = blockIdx.x * 256 + threadIdx.x;
    if (e < NE) {
        atomicAdd(&degN[ni[e]], 1.0f);
        atomicAdd(&degHE[hi[e]], 1.0f);
    }
}
__global__ void k_invert(float* p, int n) {
    int i = blockIdx.x * 256 + threadIdx.x;
    if (i < n) {
        float d = p[i];
        p[i] = (d > 0.0f) ? (1.0f / d) : 0.0f;
    }
}

// ---------- GEMM1: x1 = x @ W1   [N,128]x[128,64] via V_WMMA_F32_16X16X4_F32 ----------
// One wave per 16x16 output tile. N=100000 = 6250 row tiles; 64 cols = 4 col tiles.
__global__ void k_gemm1(const float* __restrict__ x, const float* __restrict__ W1,
                        float* __restrict__ x1) {
    int wave = (blockIdx.x * blockDim.x + threadIdx.x) >> 5;   // 8 waves / block
    int lane = threadIdx.x & 31;
    int rowTile = wave >> 2, colTile = wave & 3;
    int r0 = rowTile << 4, c0 = colTile << 4;
    int half = lane >> 4, lm = lane & 15;

    v8f acc = {};
    const float* __restrict__ arow = x + (size_t)(r0 + lm) * NF;
    for (int k0 = 0; k0 < NF; k0 += 4) {
        int k = k0 + 2 * half;                 // this lane-half holds K = k, k+1
        v2f a; a.x = arow[k]; a.y = arow[k + 1];
        v2f b; b.x = W1[(size_t)k * HC + c0 + lm];
               b.y = W1[(size_t)(k + 1) * HC + c0 + lm];
        acc = __builtin_amdgcn_wmma_f32_16x16x4_f32(false, a, false, b,
                                                    (short)0, acc, false, false);
    }
#pragma unroll
    for (int r = 0; r < 8; ++r)
        x1[(size_t)(r0 + r + 8 * half) * HC + c0 + lm] = acc[r];
}

// ---------- attention logits + segment max (by target node) ----------
__global__ void k_attn_logits(const float* __restrict__ x1, const float* __restrict__ att1,
                              const int* __restrict__ ni, const int* __restrict__ hi,
                              float* __restrict__ alphaBuf, unsigned* __restrict__ amaxEnc) {
    long long t = (long long)blockIdx.x * 256 + threadIdx.x;
    if (t >= (long long)NE * H) return;
    int e = (int)(t >> 3), h = (int)(t & 7);
    int n = ni[e], j = hi[e];
    const float* xi = x1 + (size_t)n * HC + h * C;
    const float* xj = x1 + (size_t)j * HC + h * C;   // old-PyG quirk: x1[hedge_idx]
    const float* at = att1 + h * (2 * C);
    float s = 0.0f;
#pragma unroll
    for (int c = 0; c < C; ++c) s += xi[c] * at[c] + xj[c] * at[C + c];
    s = (s > 0.0f) ? s : 0.2f * s;                   // leaky_relu(0.2)
    alphaBuf[t] = s;
    atomicMax(&amaxEnc[(size_t)n * H + h], encf(s));
}

// ---------- exp(alpha - max) + segment sum ----------
__global__ void k_attn_exp(const int* __restrict__ ni, const unsigned* __restrict__ amaxEnc,
                           float* __restrict__ alphaBuf, float* __restrict__ asum) {
    long long t = (long long)blockIdx.x * 256 + threadIdx.x;
    if (t >= (long long)NE * H) return;
    int e = (int)(t >> 3), h = (int)(t & 7);
    int n = ni[e];
    float a = expf(alphaBuf[t] - decf(amaxEnc[(size_t)n * H + h]));
    alphaBuf[t] = a;
    atomicAdd(&asum[(size_t)n * H + h], a);
}

// ---------- normalize ----------
__global__ void k_attn_norm(const int* __restrict__ ni, const float* __restrict__ asum,
                            float* __restrict__ alphaBuf) {
    long long t = (long long)blockIdx.x * 256 + threadIdx.x;
    if (t >= (long long)NE * H) return;
    int e = (int)(t >> 3), h = (int)(t & 7);
    alphaBuf[t] = alphaBuf[t] / (asum[(size_t)ni[e] * H + h] + 1e-16f);
}

// ---------- hconv layer1 stage1: nodes -> hyperedges (x alpha x Binv) ----------
__global__ void k_h1s1(const float* __restrict__ x1, const float* __restrict__ alpha,
                       const float* __restrict__ Binv, const int* __restrict__ ni,
                       const int* __restrict__ hi, float* __restrict__ he1) {
    long long t = (long long)blockIdx.x * 256 + threadIdx.x;
    if (t >= (long long)NE * HC) return;
    int e = (int)(t >> 6), c = (int)(t & 63);
    int n = ni[e], j = hi[e];
    float v = x1[(size_t)n * HC + c] * alpha[(size_t)e * H + (c >> 3)] * Binv[j];
    atomicAdd(&he1[(size_t)j * HC + c], v);
}

// ---------- hconv layer1 stage2: hyperedges -> nodes (x alpha x Dinv) ----------
__global__ void k_h1s2(const float* __restrict__ he1, const float* __restrict__ alpha,
                       const float* __restrict__ Dinv, const int* __restrict__ ni,
                       const int* __restrict__ hi, float* __restrict__ hsum) {
    long long t = (long long)blockIdx.x * 256 + threadIdx.x;
    if (t >= (long long)NE * HC) return;
    int e = (int)(t >> 6), c = (int)(t & 63);
    int n = ni[e], j = hi[e];
    float v = he1[(size_t)j * HC + c] * alpha[(size_t)e * H + (c >> 3)] * Dinv[n];
    atomicAdd(&hsum[(size_t)n * HC + c], v);
}

// ---------- pad W2 [64,8] -> [64,16] with zeros ----------
__global__ void k_padW2(const float* __restrict__ W2, float* __restrict__ W2pad) {
    int i = blockIdx.x * 256 + threadIdx.x;
    if (i < 64 * 16) {
        int k = i >> 4, c = i & 15;
        W2pad[i] = (c < C2) ? W2[k * C2 + c] : 0.0f;
    }
}

// ---------- GEMM2: x2 = elu(hsum + b1) @ W2   [N,64]x[64,8], fused bias+ELU on A ----------
__global__ void k_gemm2(const float* __restrict__ hsum, const float* __restrict__ b1,
                        const float* __restrict__ W2pad, float* __restrict__ x2) {
    int wave = (blockIdx.x * blockDim.x + threadIdx.x) >> 5;
    if (wave >= NN / 16) return;                    // uniform per wave (EXEC stays full)
    int lane = threadIdx.x & 31;
    int r0 = wave << 4;
    int half = lane >> 4, lm = lane & 15;

    v8f acc = {};
    const float* __restrict__ arow = hsum + (size_t)(r0 + lm) * HC;
    for (int k0 = 0; k0 < HC; k0 += 4) {
        int k = k0 + 2 * half;
        float a0 = arow[k] + b1[k];
        float a1 = arow[k + 1] + b1[k + 1];
        a0 = (a0 > 0.0f) ? a0 : (expf(a0) - 1.0f);  // elu
        a1 = (a1 > 0.0f) ? a1 : (expf(a1) - 1.0f);
        v2f a; a.x = a0; a.y = a1;
        v2f b; b.x = W2pad[k * 16 + lm]; b.y = W2pad[(k + 1) * 16 + lm];
        acc = __builtin_amdgcn_wmma_f32_16x16x4_f32(false, a, false, b,
                                                    (short)0, acc, false, false);
    }
    if (lm < C2) {
#pragma unroll
        for (int r = 0; r < 8; ++r)
            x2[(size_t)(r0 + r + 8 * half) * C2 + lm] = acc[r];
    }
}

// ---------- hconv layer2 (no attention), stages 1 & 2 ----------
__global__ void k_h2s1(const float* __restrict__ x2, const float* __restrict__ Binv,
                       const int* __restrict__ ni, const int* __restrict__ hi,
                       float* __restrict__ he2) {
    long long t = (long long)blockIdx.x * 256 + threadIdx.x;
    if (t >= (long long)NE * C2) return;
    int e = (int)(t >> 3), c = (int)(t & 7);
    int n = ni[e], j = hi[e];
    atomicAdd(&he2[(size_t)j * C2 + c], x2[(size_t)n * C2 + c] * Binv[j]);
}
__global__ void k_h2s2(const float* __restrict__ he2, const float* __restrict__ Dinv,
                       const int* __restrict__ ni, const int* __restrict__ hi,
                       float* __restrict__ out2) {
    long long t = (long long)blockIdx.x * 256 + threadIdx.x;
    if (t >= (long long)NE * C2) return;
    int e = (int)(t >> 3), c = (int)(t & 7);
    int n = ni[e], j = hi[e];
    atomicAdd(&out2[(size_t)n * C2 + c], he2[(size_t)j * C2 + c] * Dinv[n]);
}

// ---------- final: log_softmax(out2 + b2) ----------
__global__ void k_final(const float* __restrict__ out2, const float* __restrict__ b2,
                        float* __restrict__ out) {
    int n = blockIdx.x * 256 + threadIdx.x;
    if (n >= NN) return;
    float v[C2];
    float m = -INFINITY;
#pragma unroll
    for (int c = 0; c < C2; ++c) { v[c] = out2[(size_t)n * C2 + c] + b2[c]; m = fmaxf(m, v[c]); }
    float s = 0.0f;
#pragma unroll
    for (int c = 0; c < C2; ++c) s += expf(v[c] - m);
    float ls = logf(s);
#pragma unroll
    for (int c = 0; c < C2; ++c) out[(size_t)n * C2 + c] = v[c] - m - ls;
}

static inline int gridFor(long long n) { return (int)((n + 255) / 256); }

extern "C" void kernel_launch(void* const* d_in, const int* in_sizes, int n_in,
                              void* d_out, int out_size, void* d_ws, size_t ws_size,
                              hipStream_t stream) {
    const float* x    = (const float*)d_in[0];
    const float* W1   = (const float*)d_in[1];
    const float* att1 = (const float*)d_in[2];
    const float* b1   = (const float*)d_in[3];
    const float* W2   = (const float*)d_in[4];
    const float* b2   = (const float*)d_in[5];
    const int*   ni   = (const int*)d_in[6];
    const int*   hi   = (const int*)d_in[7];
    float* out = (float*)d_out;

    // workspace carve-up (256B aligned)
    char* w = (char*)d_ws;
    size_t used = 0;
    auto alloc = [&](size_t bytes) -> void* {
        void* p = w + used;
        used += (bytes + 255) & ~(size_t)255;
        return p;
    };
    float*    x1       = (float*)alloc((size_t)NN * HC * 4);
    float*    alphaBuf = (float*)alloc((size_t)NE * H * 4);
    unsigned* amaxEnc  = (unsigned*)alloc((size_t)NN * H * 4);
    float*    asum     = (float*)alloc((size_t)NN * H * 4);
    float*    degN     = (float*)alloc((size_t)NN * 4);          // becomes Dinv in place
    float*    degHE    = (float*)alloc((size_t)NHE * 4);         // becomes Binv in place
    float*    he1      = (float*)alloc((size_t)NHE * HC * 4);
    float*    hsum     = (float*)alloc((size_t)NN * HC * 4);
    float*    W2pad    = (float*)alloc((size_t)64 * 16 * 4);
    float*    x2       = (float*)alloc((size_t)NN * C2 * 4);
    float*    he2      = (float*)alloc((size_t)NHE * C2 * 4);
    float*    out2     = (float*)alloc((size_t)NN * C2 * 4);
    if (used > ws_size) return;  // insufficient scratch; avoid OOB

    // ---- re-init accumulators every call (graph-replay safe) ----
    k_fill_f<<<gridFor((long long)NN * H), 256, 0, stream>>>(asum, 0.0f, NN * H);
    k_fill_u<<<gridFor((long long)NN * H), 256, 0, stream>>>(amaxEnc, ENC_NEGINF, NN * H);
    k_fill_f<<<gridFor(NN), 256, 0, stream>>>(degN, 0.0f, NN);
    k_fill_f<<<gridFor(NHE), 256, 0, stream>>>(degHE, 0.0f, NHE);
    k_fill_f<<<gridFor((long long)NHE * HC), 256, 0, stream>>>(he1, 0.0f, NHE * HC);
    k_fill_f<<<gridFor((long long)NN * HC), 256, 0, stream>>>(hsum, 0.0f, NN * HC);
    k_fill_f<<<gridFor((long long)NHE * C2), 256, 0, stream>>>(he2, 0.0f, NHE * C2);
    k_fill_f<<<gridFor((long long)NN * C2), 256, 0, stream>>>(out2, 0.0f, NN * C2);

    // ---- degrees -> reciprocals ----
    k_degree<<<gridFor(NE), 256, 0, stream>>>(ni, hi, degN, degHE);
    k_invert<<<gridFor(NN), 256, 0, stream>>>(degN, NN);
    k_invert<<<gridFor(NHE), 256, 0, stream>>>(degHE, NHE);

    // ---- GEMM1 via WMMA f32 16x16x4: 6250*4 tiles, 8 waves/block ----
    k_gemm1<<<(NN / 16) * 4 / 8, 256, 0, stream>>>(x, W1, x1);

    // ---- attention softmax (segmented by target node) ----
    k_attn_logits<<<gridFor((long long)NE * H), 256, 0, stream>>>(x1, att1, ni, hi, alphaBuf, amaxEnc);
    k_attn_exp   <<<gridFor((long long)NE * H), 256, 0, stream>>>(ni, amaxEnc, alphaBuf, asum);
    k_attn_norm  <<<gridFor((long long)NE * H), 256, 0, stream>>>(ni, asum, alphaBuf);

    // ---- hconv layer 1 ----
    k_h1s1<<<gridFor((long long)NE * HC), 256, 0, stream>>>(x1, alphaBuf, degHE, ni, hi, he1);
    k_h1s2<<<gridFor((long long)NE * HC), 256, 0, stream>>>(he1, alphaBuf, degN, ni, hi, hsum);

    // ---- GEMM2 via WMMA (fused +b1, ELU on A operand) ----
    k_padW2<<<gridFor(64 * 16), 256, 0, stream>>>(W2, W2pad);
    k_gemm2<<<(NN / 16 + 7) / 8, 256, 0, stream>>>(hsum, b1, W2pad, x2);

    // ---- hconv layer 2 (no attention) ----
    k_h2s1<<<gridFor((long long)NE * C2), 256, 0, stream>>>(x2, degHE, ni, hi, he2);
    k_h2s2<<<gridFor((long long)NE * C2), 256, 0, stream>>>(he2, degN, ni, hi, out2);

    // ---- log_softmax ----
    k_final<<<gridFor(NN), 256, 0, stream>>>(out2, b2, out);
}